// RouteActivation_11235634446548
// MI455X (gfx1250) — compile-verified
//
#include <hip/hip_runtime.h>
#include <math.h>
#include <stdint.h>

// ---------------------------------------------------------------------------
// RouteActivation: fused LayerNorm(D=2048) + Linear(D->1) + sigmoid/T + clip
// Memory-bound (1.07 GB read @ 23.3 TB/s => ~46us floor). Single pass over x
// with 3 streaming reductions per row. No WMMA on purpose: the A/B lane
// layouts would stride lanes across rows 8KB apart (kills coalescing) and
// cannot express sum(x^2) without an extra elementwise pass; VALU demand at
// full HBM rate (~35 TFLOP/s f32) is far below VALU peak, so FMA reductions
// keep the kernel bandwidth-bound.
// CDNA5 paths: async global->LDS staging of parameters (ASYNCcnt),
// s_wait_asynccnt, non-temporal b128 streaming loads, wave32 shuffles.
// ---------------------------------------------------------------------------

#define DD 2048
#define NROWS 131072
#define TEMP_INV (1.0f / 0.7f)
#define GATE_MIN 0.02f
#define GATE_MAX 0.98f
#define LN_EPS 1e-5f

#define WAVE 32
#define WAVES_PER_BLOCK 8
#define NTHREADS (WAVES_PER_BLOCK * WAVE)            // 256
#define NBLOCKS 2048
#define ROWS_PER_BLOCK (NROWS / NBLOCKS)             // 64
#define ROW_ITERS (ROWS_PER_BLOCK / WAVES_PER_BLOCK) // 8
#define ELEMS_PER_THREAD (DD / NTHREADS)             // 8
#define VEC_PER_LANE (DD / (WAVE * 4))               // 16 x float4 per lane/row

typedef float f32x4 __attribute__((ext_vector_type(4)));

// The async-LDS builtin takes GCC-style int vectors of 4 ("V4i") in AS(1)
// (global, prints as __device__ in HIP diagnostics) and AS(3) (LDS).
typedef int i32x4 __attribute__((vector_size(16)));
typedef __attribute__((address_space(1))) i32x4 as1_i32x4;
typedef __attribute__((address_space(3))) i32x4 as3_i32x4;

// 16B async DMA global -> LDS (CDNA5 GLOBAL_LOAD_ASYNC_TO_LDS_B128).
// Generic->LDS offset: addrspacecast generic(shared)->local is a truncation,
// so the low 32 bits of the generic shared pointer are the LDS byte offset.
__device__ __forceinline__ void async_copy_b128(const void* gsrc, void* ldst) {
#if __has_builtin(__builtin_amdgcn_global_load_async_to_lds_b128)
  __builtin_amdgcn_global_load_async_to_lds_b128(
      (as1_i32x4*)(uintptr_t)gsrc,
      (as3_i32x4*)(uint32_t)(uintptr_t)ldst,
      /*offset=*/0, /*cpol=*/0);
#else
  *(f32x4*)ldst = *(const f32x4*)gsrc; // fallback: regular load + ds_store
#endif
}

__device__ __forceinline__ void wait_async_lds() {
#if __has_builtin(__builtin_amdgcn_s_wait_asynccnt)
  __builtin_amdgcn_s_wait_asynccnt(0);
#else
  asm volatile("s_wait_asynccnt 0x0" ::: "memory");
#endif
}

__global__ __launch_bounds__(NTHREADS) void route_gate_kernel(
    const float* __restrict__ x, const float* __restrict__ lnw,
    const float* __restrict__ lnb, const float* __restrict__ w,
    const float* __restrict__ bias, float* __restrict__ out) {
  __shared__ __align__(16) float sGW[DD]; // staged gamma, then gamma*w in place
  __shared__ __align__(16) float sW[DD];
  __shared__ __align__(16) float sB[DD];
  __shared__ float sRed1[WAVES_PER_BLOCK];
  __shared__ float sRed2[WAVES_PER_BLOCK];
  __shared__ float sC[2];

  const int tid = threadIdx.x;
  const int wave = tid >> 5;
  const int lane = tid & 31;

  // ---- Stage ln_weight / w / ln_bias into LDS via async DMA (once/block) ----
#pragma unroll
  for (int i = 0; i < ELEMS_PER_THREAD / 4; ++i) { // 2 x b128 per thread/array
    const int idx = (i * NTHREADS + tid) * 4;
    async_copy_b128(lnw + idx, sGW + idx);
    async_copy_b128(w + idx, sW + idx);
    async_copy_b128(lnb + idx, sB + idx);
  }
  wait_async_lds();
  __syncthreads();

  // ---- gw[j] = gamma[j]*w[j] in place; C1 = sum(gw); C2 = sum(beta*w)+b ----
  float c1 = 0.f, c2 = 0.f;
#pragma unroll
  for (int i = 0; i < ELEMS_PER_THREAD; ++i) {
    const int j = i * NTHREADS + tid;
    const float gv = sGW[j], wv = sW[j];
    const float p = gv * wv;
    sGW[j] = p;
    c1 += p;
    c2 = fmaf(sB[j], wv, c2);
  }
#pragma unroll
  for (int m = 16; m > 0; m >>= 1) {
    c1 += __shfl_xor(c1, m, WAVE);
    c2 += __shfl_xor(c2, m, WAVE);
  }
  if (lane == 0) {
    sRed1[wave] = c1;
    sRed2[wave] = c2;
  }
  __syncthreads();
  if (tid == 0) {
    float a = 0.f, c = 0.f;
    for (int i = 0; i < WAVES_PER_BLOCK; ++i) {
      a += sRed1[i];
      c += sRed2[i];
    }
    sC[0] = a;
    sC[1] = c + bias[0];
  }
  __syncthreads();
  const float C1 = sC[0];
  const float C2 = sC[1];

  const f32x4* gw4 = (const f32x4*)sGW;

  // ---- Stream rows: one wave per row, contiguous 512B per load instr ----
#pragma unroll 1
  for (int it = 0; it < ROW_ITERS; ++it) {
    const int row = blockIdx.x * ROWS_PER_BLOCK + it * WAVES_PER_BLOCK + wave;
    const f32x4* xr = (const f32x4*)(x + (size_t)row * DD);
    float s1 = 0.f, s2 = 0.f, s3 = 0.f;
#pragma unroll
    for (int i = 0; i < VEC_PER_LANE; ++i) {
      const f32x4 v = __builtin_nontemporal_load(&xr[i * WAVE + lane]);
      const f32x4 g = gw4[i * WAVE + lane];
      s1 += (v.x + v.y) + (v.z + v.w);
      s2 = fmaf(v.x, v.x, fmaf(v.y, v.y, fmaf(v.z, v.z, fmaf(v.w, v.w, s2))));
      s3 = fmaf(v.x, g.x, fmaf(v.y, g.y, fmaf(v.z, g.z, fmaf(v.w, g.w, s3))));
    }
#pragma unroll
    for (int m = 16; m > 0; m >>= 1) {
      s1 += __shfl_xor(s1, m, WAVE);
      s2 += __shfl_xor(s2, m, WAVE);
      s3 += __shfl_xor(s3, m, WAVE);
    }
    if (lane == 0) {
      const float mu = s1 * (1.0f / DD);
      const float var = fmaf(-mu, mu, s2 * (1.0f / DD));
      const float rstd = rsqrtf(var + LN_EPS);
      const float z = fmaf(rstd, s3 - mu * C1, C2) * TEMP_INV;
      float a = 1.0f / (1.0f + __expf(-z));
      a = fminf(fmaxf(a, GATE_MIN), GATE_MAX);
      out[row] = a;
    }
  }
}

extern "C" void kernel_launch(void* const* d_in, const int* in_sizes, int n_in,
                              void* d_out, int out_size, void* d_ws,
                              size_t ws_size, hipStream_t stream) {
  const float* x = (const float*)d_in[0];
  const float* lnw = (const float*)d_in[1];
  const float* lnb = (const float*)d_in[2];
  const float* w = (const float*)d_in[3];
  const float* b = (const float*)d_in[4];
  float* out = (float*)d_out;

  route_gate_kernel<<<NBLOCKS, NTHREADS, 0, stream>>>(x, lnw, lnb, w, b, out);

  (void)in_sizes;
  (void)n_in;
  (void)out_size;
  (void)d_ws;
  (void)ws_size;
}